// BioGNN_14551349199581
// MI455X (gfx1250) — compile-verified
//
#include <hip/hip_runtime.h>

// BioGNN edge-scatter update for MI455X (gfx1250, wave32).
// Memory-bound scatter/gather (segment-sum): no dense matrix structure -> no WMMA.
// CDNA5 paths used:
//   - GLOBAL_LOAD_ASYNC_TO_LDS_B128 + s_wait_asynccnt: double-buffered edge stream
//   - global_atomic_add_f32 (non-returning, L2 atomic units)
//   - global_load_b128 / ds_load_b128 / global_store_b8

#define TILE_EDGES 1024   // edges per tile per block (256 thr x 4 edges)

typedef int v4i __attribute__((ext_vector_type(4)));
typedef __attribute__((address_space(1))) v4i gas_v4i;   // global (device) AS
typedef __attribute__((address_space(3))) v4i las_v4i;   // LDS (shared) AS

__device__ __forceinline__ void fadd_atomic(float* p, float v) {
  unsafeAtomicAdd(p, v);  // HW FP32 atomic add, non-returning form
}

__device__ __forceinline__ void compiler_fence() { asm volatile("" ::: "memory"); }

// Async copy of 16 bytes global->LDS (per lane). ASYNCcnt path.
__device__ __forceinline__ void async_copy_b128(const void* g, void* l) {
#if defined(__has_builtin) && __has_builtin(__builtin_amdgcn_global_load_async_to_lds_b128)
  // AS1 ptr: flat/global share numeric addresses. AS3 ptr: low 32 bits of a
  // flat LDS address are the LDS offset.
  __builtin_amdgcn_global_load_async_to_lds_b128(
      (gas_v4i*)(uintptr_t)g,
      (las_v4i*)(unsigned)(uintptr_t)l,
      0, 0);
#else
  const unsigned lo = (unsigned)(uintptr_t)l;
  asm volatile("global_load_async_to_lds_b128 %0, %1, off" :: "v"(lo), "v"(g) : "memory");
#endif
}

__device__ __forceinline__ void wait_asynccnt_le3() {
#if defined(__has_builtin) && __has_builtin(__builtin_amdgcn_s_wait_asynccnt)
  __builtin_amdgcn_s_wait_asynccnt(3);
#else
  asm volatile("s_wait_asynccnt 0x3" ::: "memory");
#endif
}

__device__ __forceinline__ void wait_asynccnt_zero() {
#if defined(__has_builtin) && __has_builtin(__builtin_amdgcn_s_wait_asynccnt)
  __builtin_amdgcn_s_wait_asynccnt(0);
#else
  asm volatile("s_wait_asynccnt 0x0" ::: "memory");
#endif
}

__global__ void __launch_bounds__(256)
bio_init(float* __restrict__ act_sum, float* __restrict__ inh_sum,
         unsigned char* __restrict__ has_act, unsigned char* __restrict__ has_inh, int n)
{
  const int g = blockIdx.x * blockDim.x + threadIdx.x;
  const int base = g << 2;
  if (base + 3 < n) {
    *reinterpret_cast<float4*>(act_sum + base) = make_float4(0.f, 0.f, 0.f, 0.f);
    *reinterpret_cast<float4*>(inh_sum + base) = make_float4(0.f, 0.f, 0.f, 0.f);
    *reinterpret_cast<unsigned int*>(has_act + base) = 0u;
    *reinterpret_cast<unsigned int*>(has_inh + base) = 0u;
  } else {
    for (int i = base; i < n; ++i) {
      act_sum[i] = 0.f; inh_sum[i] = 0.f; has_act[i] = 0; has_inh[i] = 0;
    }
  }
}

__global__ void __launch_bounds__(256)
bio_edges(const float* __restrict__ x,
          const float* __restrict__ k_act, const int* __restrict__ act_src, const int* __restrict__ act_dst,
          const float* __restrict__ k_inh, const int* __restrict__ inh_src, const int* __restrict__ inh_dst,
          float* __restrict__ act_sum, float* __restrict__ inh_sum,
          unsigned char* __restrict__ has_act, unsigned char* __restrict__ has_inh,
          int n_act, int n_inh)
{
  __shared__ alignas(16) int   ls_src[2][TILE_EDGES];
  __shared__ alignas(16) int   ls_dst[2][TILE_EDGES];
  __shared__ alignas(16) float ls_k  [2][TILE_EDGES];

  const int e0 = (int)threadIdx.x << 2;               // this thread's 4-edge slice
  const unsigned ta = (unsigned)n_act / TILE_EDGES;   // full act tiles
  const unsigned ti = (unsigned)n_inh / TILE_EDGES;   // full inh tiles
  const unsigned ttot = ta + ti;

  // Issue one tile's async loads: each thread copies its own 16B of src/dst/k.
  auto issue = [&](unsigned tileIdx, int b) {
    const int* src; const int* dst; const float* k; int base;
    if (tileIdx < ta) { src = act_src; dst = act_dst; k = k_act; base = (int)tileIdx * TILE_EDGES; }
    else              { src = inh_src; dst = inh_dst; k = k_inh; base = (int)(tileIdx - ta) * TILE_EDGES; }
    async_copy_b128(src + base + e0, &ls_src[b][e0]);
    async_copy_b128(dst + base + e0, &ls_dst[b][e0]);
    async_copy_b128(k   + base + e0, &ls_k  [b][e0]);
  };

  // Process one tile: read own slice from LDS, gather x, scatter-add + flags.
  auto process = [&](unsigned tileIdx, int b) {
    float* sum; unsigned char* has;
    if (tileIdx < ta) { sum = act_sum; has = has_act; }
    else              { sum = inh_sum; has = has_inh; }
    const int4   s4 = *reinterpret_cast<const int4*>(&ls_src[b][e0]);
    const int4   d4 = *reinterpret_cast<const int4*>(&ls_dst[b][e0]);
    const float4 k4 = *reinterpret_cast<const float4*>(&ls_k[b][e0]);
    const float v0 = x[s4.x];
    const float v1 = x[s4.y];
    const float v2 = x[s4.z];
    const float v3 = x[s4.w];
    fadd_atomic(&sum[d4.x], k4.x * v0 * v0); has[d4.x] = 1;
    fadd_atomic(&sum[d4.y], k4.y * v1 * v1); has[d4.y] = 1;
    fadd_atomic(&sum[d4.z], k4.z * v2 * v2); has[d4.z] = 1;
    fadd_atomic(&sum[d4.w], k4.w * v3 * v3); has[d4.w] = 1;
  };

  // Tiles for this block: blockIdx.x + i*gridDim.x, double-buffered pipeline.
  unsigned nblk = 0;
  if (blockIdx.x < ttot) nblk = (ttot - 1u - blockIdx.x) / gridDim.x + 1u;

  if (nblk) {
    issue(blockIdx.x, 0);
    for (unsigned i = 0; i < nblk; ++i) {
      const unsigned cur = blockIdx.x + i * gridDim.x;
      const int b = (int)(i & 1u);
      if (i + 1u < nblk) { issue(cur + gridDim.x, b ^ 1); wait_asynccnt_le3(); }
      else               { wait_asynccnt_zero(); }
      compiler_fence();
      process(cur, b);
      compiler_fence();
    }
  }

  // Tail edges (n_e % TILE_EDGES) via direct path, distributed over whole grid.
  const int gsz = (int)(gridDim.x * blockDim.x);
  const int gid = (int)(blockIdx.x * blockDim.x + threadIdx.x);
  for (int e = (int)(ta * TILE_EDGES) + gid; e < n_act; e += gsz) {
    const float v = x[act_src[e]];
    fadd_atomic(&act_sum[act_dst[e]], k_act[e] * v * v);
    has_act[act_dst[e]] = 1;
  }
  for (int e = (int)(ti * TILE_EDGES) + gid; e < n_inh; e += gsz) {
    const float v = x[inh_src[e]];
    fadd_atomic(&inh_sum[inh_dst[e]], k_inh[e] * v * v);
    has_inh[inh_dst[e]] = 1;
  }
}

__device__ __forceinline__ float node_update(float as, float is,
                                             unsigned char fa, unsigned char fi,
                                             float xv, float nuv, float dec, float gro)
{
  const float num = fa ? as : 1.0f;
  const float den = 1.0f + is;
  const float agg = (fa | fi) ? (num / den) : 0.0f;
  return nuv * agg - dec * xv + gro;
}

__global__ void __launch_bounds__(256)
bio_finalize(const float* __restrict__ x,  const float* __restrict__ nu,
             const float* __restrict__ decay, const float* __restrict__ growth,
             const float* __restrict__ act_sum, const float* __restrict__ inh_sum,
             const unsigned char* __restrict__ has_act, const unsigned char* __restrict__ has_inh,
             float* __restrict__ out, int n)
{
  const int g = blockIdx.x * blockDim.x + threadIdx.x;
  const int base = g << 2;
  if (base + 3 < n) {
    const float4 as = *reinterpret_cast<const float4*>(act_sum + base);
    const float4 is = *reinterpret_cast<const float4*>(inh_sum + base);
    const uchar4 fa = *reinterpret_cast<const uchar4*>(has_act + base);
    const uchar4 fi = *reinterpret_cast<const uchar4*>(has_inh + base);
    const float4 xv = *reinterpret_cast<const float4*>(x + base);
    const float4 nv = *reinterpret_cast<const float4*>(nu + base);
    const float4 dv = *reinterpret_cast<const float4*>(decay + base);
    const float4 gv = *reinterpret_cast<const float4*>(growth + base);
    float4 r;
    r.x = node_update(as.x, is.x, fa.x, fi.x, xv.x, nv.x, dv.x, gv.x);
    r.y = node_update(as.y, is.y, fa.y, fi.y, xv.y, nv.y, dv.y, gv.y);
    r.z = node_update(as.z, is.z, fa.z, fi.z, xv.z, nv.z, dv.z, gv.z);
    r.w = node_update(as.w, is.w, fa.w, fi.w, xv.w, nv.w, dv.w, gv.w);
    *reinterpret_cast<float4*>(out + base) = r;
  } else {
    for (int i = base; i < n; ++i) {
      out[i] = node_update(act_sum[i], inh_sum[i], has_act[i], has_inh[i],
                           x[i], nu[i], decay[i], growth[i]);
    }
  }
}

extern "C" void kernel_launch(void* const* d_in, const int* in_sizes, int n_in,
                              void* d_out, int out_size, void* d_ws, size_t ws_size,
                              hipStream_t stream)
{
  const float* x      = (const float*)d_in[0];
  const float* k_act  = (const float*)d_in[1];
  const float* k_inh  = (const float*)d_in[2];
  const float* nu     = (const float*)d_in[3];
  const float* decay  = (const float*)d_in[4];
  const float* growth = (const float*)d_in[5];
  const int* act_src  = (const int*)d_in[6];
  const int* act_dst  = (const int*)d_in[7];
  const int* inh_src  = (const int*)d_in[8];
  const int* inh_dst  = (const int*)d_in[9];

  const int n     = in_sizes[0];
  const int n_act = in_sizes[6];
  const int n_inh = in_sizes[8];

  // Workspace layout: [act_sum f32 x n][inh_sum f32 x n][has_act u8 x n][has_inh u8 x n]
  float* act_sum = (float*)d_ws;
  float* inh_sum = act_sum + n;
  unsigned char* has_act = (unsigned char*)(inh_sum + n);
  unsigned char* has_inh = has_act + n;

  const int T = 256;                       // 8 wave32 waves per block
  const int grp_n = (n + 3) / 4;
  const int node_blocks = (grp_n + T - 1) / T;

  bio_init<<<node_blocks, T, 0, stream>>>(act_sum, inh_sum, has_act, has_inh, n);

  const unsigned tiles = (unsigned)(n_act / TILE_EDGES) + (unsigned)(n_inh / TILE_EDGES);
  unsigned blocks = 4096;                  // ~1M threads; 4 tiles per block here
  if (tiles > 0 && blocks > tiles) blocks = tiles;
  if (blocks == 0) blocks = 1;
  bio_edges<<<blocks, T, 0, stream>>>(x, k_act, act_src, act_dst,
                                      k_inh, inh_src, inh_dst,
                                      act_sum, inh_sum, has_act, has_inh,
                                      n_act, n_inh);

  bio_finalize<<<node_blocks, T, 0, stream>>>(x, nu, decay, growth,
                                              act_sum, inh_sum, has_act, has_inh,
                                              (float*)d_out, n);
}